// LIFCell_3547642987405
// MI455X (gfx1250) — compile-verified
//
#include <hip/hip_runtime.h>
#include <math.h>
#include <stdint.h>

// LIF scan: v = v*decay*(1-z) + x_t ; z = (v - 0.3 > 0)
// B=128, T=2048, H=128 fp32. Memory-bound: 268 MB @ 23.3 TB/s ~ 11.5us floor.
// Parallelism is capped at B*H = 512 waves by the serial scan, so HBM latency
// is hidden structurally: quad-buffered LDS pipeline (depth-3 prefetch) fed by
// the CDNA5 async global->LDS data mover (ASYNCcnt + split barriers). In-flight
// ~12 MB of loads device-wide + concurrent NT store stream. Both global
// streams are touch-once (134 MB each > 192 MB L2 combined) -> NT hints.

#define V_THRESH    0.3f
#define H_DIM       128
#define CHUNK_T     64
#define CHUNK_FLOATS (CHUNK_T * H_DIM)            // 8192 floats = 32 KB
#define CHUNK_BYTES  (CHUNK_FLOATS * 4)
#define NBUF        4                              // 128 KB LDS: 2 blocks/WGP
#define VEC_PER_THR 16                             // 16 x b128 per thread per chunk

// INST_OFFSET is added to BOTH the LDS destination and the global source
// (CDNA5 ISA), and within a chunk both advance in lockstep
// (+2048 B per issue: 128 threads * 16 B). So one LDS-addr VGPR + one global
// base pair + immediate offsets covers the whole chunk with zero VALU adds.
#define ASYNC_LD128(off_lit)                                                   \
  asm volatile("global_load_async_to_lds_b128 %0, %1, off offset:" #off_lit    \
               " th:TH_LOAD_NT"                                                \
               :: "v"(l), "v"(g) : "memory")

__device__ __forceinline__ void issue_chunk_async(unsigned l, unsigned long long g) {
  ASYNC_LD128(0);
  ASYNC_LD128(2048);
  ASYNC_LD128(4096);
  ASYNC_LD128(6144);
  ASYNC_LD128(8192);
  ASYNC_LD128(10240);
  ASYNC_LD128(12288);
  ASYNC_LD128(14336);
  ASYNC_LD128(16384);
  ASYNC_LD128(18432);
  ASYNC_LD128(20480);
  ASYNC_LD128(22528);
  ASYNC_LD128(24576);
  ASYNC_LD128(26624);
  ASYNC_LD128(28672);
  ASYNC_LD128(30720);
}

__global__ __launch_bounds__(H_DIM) void lif_scan_async_kernel(
    const float* __restrict__ x_seq,     // [B, T, H]
    const float* __restrict__ v0,        // [B, H]
    const float* __restrict__ z0,        // [B, H]
    const float* __restrict__ decay_raw, // [H]
    float* __restrict__ out,             // [B, T, H]
    int T) {
  __shared__ float smem[NBUF][CHUNK_FLOATS];

  const int h = threadIdx.x;
  const int b = blockIdx.x;

  const float decay = 1.0f / (1.0f + __expf(-decay_raw[h]));
  float v = v0[b * H_DIM + h];
  float z = z0[b * H_DIM + h];

  const size_t row_base = (size_t)b * (size_t)T * H_DIM;
  const unsigned long long xbase =
      (unsigned long long)(uintptr_t)x_seq + (unsigned long long)row_base * 4ull +
      (unsigned long long)threadIdx.x * 16ull;
  float* outp = out + row_base + h;

  const int nc = T / CHUNK_T;

  // LDS byte offsets (low 32 bits of flat LDS address) + per-thread lane slot.
  unsigned lbuf[NBUF];
#pragma unroll
  for (int i = 0; i < NBUF; ++i)
    lbuf[i] = (unsigned)(uintptr_t)(&smem[i][0]) + threadIdx.x * 16u;

  // Prologue: prefetch chunks 0..2.
  if (nc > 0) issue_chunk_async(lbuf[0], xbase);
  if (nc > 1) issue_chunk_async(lbuf[1], xbase + 1ull * CHUNK_BYTES);
  if (nc > 2) issue_chunk_async(lbuf[2], xbase + 2ull * CHUNK_BYTES);

  for (int c = 0; c < nc; ++c) {
    // Keep three chunks in flight; ASYNCcnt retires in order, so waiting down
    // to <= 3*VEC_PER_THR outstanding guarantees chunk c has fully landed.
    if (c + 3 < nc) {
      issue_chunk_async(lbuf[(c + 3) % NBUF],
                        xbase + (unsigned long long)(c + 3) * CHUNK_BYTES);
      asm volatile("s_wait_asynccnt 48" ::: "memory");  // 3 * VEC_PER_THR
    } else if (c + 2 < nc) {
      asm volatile("s_wait_asynccnt 32" ::: "memory");  // 2 * VEC_PER_THR
    } else if (c + 1 < nc) {
      asm volatile("s_wait_asynccnt 16" ::: "memory");  // 1 * VEC_PER_THR
    } else {
      asm volatile("s_wait_asynccnt 0" ::: "memory");
    }
    __syncthreads();  // all waves' DMA for chunk c visible to whole workgroup

    const float* buf = &smem[c % NBUF][0];
    float* op = outp + (size_t)c * CHUNK_T * H_DIM;
#pragma unroll 8
    for (int i = 0; i < CHUNK_T; ++i) {
      const float x = buf[i * H_DIM + h];        // ds_load_2addr_stride64 pairs
      v = v * decay * (1.0f - z) + x;
      z = (v > V_THRESH) ? 1.0f : 0.0f;
      __builtin_nontemporal_store(z, &op[(size_t)i * H_DIM]);  // coalesced NT
    }
    __syncthreads();  // buffer (c%NBUF) free before it is re-filled
  }

  // Tail (T not divisible by CHUNK_T): direct global reads.
  for (int t = nc * CHUNK_T; t < T; ++t) {
    const float x = x_seq[row_base + (size_t)t * H_DIM + h];
    v = v * decay * (1.0f - z) + x;
    z = (v > V_THRESH) ? 1.0f : 0.0f;
    __builtin_nontemporal_store(z, &out[row_base + (size_t)t * H_DIM + h]);
  }
}

// Generic fallback for H != 128 (one thread per (b,h), direct streaming).
__global__ void lif_scan_generic_kernel(const float* __restrict__ x_seq,
                                        const float* __restrict__ v0,
                                        const float* __restrict__ z0,
                                        const float* __restrict__ decay_raw,
                                        float* __restrict__ out,
                                        int B, int T, int H) {
  int idx = blockIdx.x * blockDim.x + threadIdx.x;
  if (idx >= B * H) return;
  const int b = idx / H, h = idx % H;
  const float decay = 1.0f / (1.0f + __expf(-decay_raw[h]));
  float v = v0[idx], z = z0[idx];
  const size_t base = (size_t)b * T * H + h;
  for (int t = 0; t < T; ++t) {
    const float x = x_seq[base + (size_t)t * H];
    v = v * decay * (1.0f - z) + x;
    z = (v > V_THRESH) ? 1.0f : 0.0f;
    __builtin_nontemporal_store(z, &out[base + (size_t)t * H]);
  }
}

extern "C" void kernel_launch(void* const* d_in, const int* in_sizes, int n_in,
                              void* d_out, int out_size, void* d_ws, size_t ws_size,
                              hipStream_t stream) {
  const float* x_seq     = (const float*)d_in[0];
  const float* v0        = (const float*)d_in[1];
  const float* z0        = (const float*)d_in[2];
  const float* decay_raw = (const float*)d_in[3];
  float* out             = (float*)d_out;

  const int H = in_sizes[3];                 // 128
  const int B = in_sizes[1] / H;             // 128
  const int T = in_sizes[0] / (B * H);       // 2048

  if (H == H_DIM) {
    lif_scan_async_kernel<<<dim3(B), dim3(H_DIM), 0, stream>>>(
        x_seq, v0, z0, decay_raw, out, T);
  } else {
    const int n = B * H;
    lif_scan_generic_kernel<<<dim3((n + 255) / 256), dim3(256), 0, stream>>>(
        x_seq, v0, z0, decay_raw, out, B, T, H);
  }
}